// DFHLoss_35940286333615
// MI455X (gfx1250) — compile-verified
//
#include <hip/hip_runtime.h>
#include <hip/hip_bf16.h>
#include <math.h>

#define TRAIN  200000
#define BATCH  256
#define NBIT   64
#define NCLASS 100
#define NTILES (TRAIN / 16)   // 12500
#define NBLK   250            // blocks for the fused GEMM+loss kernel

typedef float v2f __attribute__((ext_vector_type(2)));
typedef float v8f __attribute__((ext_vector_type(8)));

__device__ __forceinline__ float signf0(float v) {
    return (v > 0.f) ? 1.f : ((v < 0.f) ? -1.f : 0.f);
}

// ---------------- small prep kernels ----------------

__global__ void k_init_mark(int* mark) {
    int t = blockIdx.x * 256 + threadIdx.x;
    if (t < TRAIN) mark[t] = 0;
}

__global__ void k_labels(const float* __restrict__ y, int* __restrict__ label) {
    int j = threadIdx.x;                 // 1 block x 256
    int lab = 0;
    for (int c = 0; c < NCLASS; ++c) {
        if (y[j * NCLASS + c] > 0.5f) { lab = c; break; }
    }
    label[j] = lab;
}

__global__ void k_blabel(const float* __restrict__ Y, int* __restrict__ blabel) {
    int t = blockIdx.x * 256 + threadIdx.x;
    if (t >= TRAIN) return;
    int lab = 0;
    for (int c = 0; c < NCLASS; ++c) {
        if (Y[(size_t)c * TRAIN + t] > 0.5f) { lab = c; break; }
    }
    blabel[t] = lab;
}

__global__ void k_scatter(const int* __restrict__ ind, const int* __restrict__ label,
                          int* __restrict__ mark, int* __restrict__ blabel) {
    int j = threadIdx.x;                 // 1 block x 256
    int t = ind[j];
    mark[t]   = j + 1;                   // column t of U_new comes from u row j
    blabel[t] = label[j];                // Y_new scatter collapses to label scatter
}

// b[k][j] = sign(MU * (C @ y^T)[k,j] + u[j,k]);  quant partial = sum (b - u^T)^2 per k-row
__global__ void k_bq(const float* __restrict__ C, const float* __restrict__ u,
                     const float* __restrict__ y, float* __restrict__ b,
                     float* __restrict__ quantP) {
    int k = blockIdx.x;                  // 64 blocks
    int j = threadIdx.x;                 // 256 threads
    float dot = 0.f;
    for (int c = 0; c < NCLASS; ++c) dot += C[k * NCLASS + c] * y[j * NCLASS + c];
    float uv = u[j * NBIT + k];
    float bv = signf0(dot + uv);         // MU = 1
    b[k * BATCH + j] = bv;
    float d = bv - uv;
    __shared__ float red[256];
    red[j] = d * d;
    __syncthreads();
    for (int s = 128; s > 0; s >>= 1) {
        if (j < s) red[j] += red[j + s];
        __syncthreads();
    }
    if (j == 0) quantP[k] = red[0];
}

__global__ void k_Q(const float* __restrict__ y, const float* __restrict__ b,
                    float* __restrict__ Q) {
    int idx = blockIdx.x * 256 + threadIdx.x;   // NCLASS*NBIT = 6400
    if (idx >= NCLASS * NBIT) return;
    int i = idx / NBIT, k = idx % NBIT;
    float s = 0.f;
    for (int j = 0; j < BATCH; ++j) s += y[j * NCLASS + i] * b[k * BATCH + j];
    Q[i * NBIT + k] = s;
}

__global__ void k_G(const float* __restrict__ y, float* __restrict__ G) {
    int idx = blockIdx.x * 256 + threadIdx.x;   // NCLASS*NCLASS = 10000
    if (idx >= NCLASS * NCLASS) return;
    int i = idx / NCLASS, i2 = idx % NCLASS;
    float s = 0.f;
    for (int j = 0; j < BATCH; ++j) s += y[j * NCLASS + i] * y[j * NCLASS + i2];
    G[idx] = s;
}

// Sequential discrete cyclic coordinate descent; 1 block x 64 threads (thread = bit k)
__global__ void k_dcc(const float* __restrict__ C, const float* __restrict__ Q,
                      const float* __restrict__ G, float* __restrict__ out) {
    __shared__ float Ct[NCLASS * NBIT];  // [class][bit]
    int k = threadIdx.x;
    float tot = 0.f;
    for (int i = 0; i < NCLASS; ++i) {
        float v = C[k * NCLASS + i];
        Ct[i * NBIT + k] = v;
        tot += v;
    }
    __syncthreads();
    for (int i = 0; i < NCLASS; ++i) {
        float dot = 0.f;
        for (int i2 = 0; i2 < NCLASS; ++i2)
            if (i2 != i) dot += Ct[i2 * NBIT + k] * G[i2 * NCLASS + i];
        float old = Ct[i * NBIT + k];
        float v   = Q[i * NBIT + k] - dot - 1.0f * (tot - old);   // VUL = 1
        float nr  = signf0(v);
        tot += nr - old;
        Ct[i * NBIT + k] = nr;
        __syncthreads();
    }
    for (int i = 0; i < NCLASS; ++i)
        out[1 + k * NCLASS + i] = Ct[i * NBIT + k];  // C_new (nbit, nclass) row-major
}

// ---------------- fused WMMA GEMM + logistic loss ----------------
// Each wave owns a 16-column N-tile of U (with virtual scatter via mark[]),
// loops all 16 M-tiles of u, and folds softplus terms into a running sum.
__global__ __launch_bounds__(256) void k_gemm_loss(
        const float* __restrict__ u, const float* __restrict__ U,
        const int* __restrict__ mark, const int* __restrict__ blabel,
        const int* __restrict__ label, float* __restrict__ metricP) {
    __shared__ float su[BATCH * 65];     // u staged row-major, pitch 65 (bank-conflict-free)
    __shared__ int   slab[BATCH];
    __shared__ float red[256];
    int tid = threadIdx.x;
    for (int x = tid; x < BATCH * NBIT; x += 256) {
        int r = x >> 6, c = x & 63;
        su[r * 65 + c] = u[x];
    }
    slab[tid] = label[tid];
    __syncthreads();

    int lane    = tid & 31;
    int wid     = tid >> 5;
    int halfsel = lane >> 4;             // 0: lanes 0-15, 1: lanes 16-31
    int nlocal  = lane & 15;
    int krow    = halfsel * 2;           // K sub-row pair per ISA layout
    int gwave   = blockIdx.x * 8 + wid;
    int gstride = gridDim.x * 8;

    float partial = 0.f;
    for (int tile = gwave; tile < NTILES; tile += gstride) {
        int n0  = tile * 16;
        int myn = n0 + nlocal;
        int mk  = mark[myn];
        int bl  = blabel[myn];

        v2f bfrag[16];                   // whole 64x16 B block: K-step kb -> B[4kb..4kb+3][n0..n0+15]
        #pragma unroll
        for (int kb = 0; kb < 16; ++kb) {
            int k = kb * 4 + krow;
            float bx, by;
            if (mk > 0) {                // overridden column: U_new[:,myn] = u[mk-1,:]
                bx = u[(mk - 1) * NBIT + k];
                by = u[(mk - 1) * NBIT + k + 1];
            } else {
                bx = U[(size_t)k * TRAIN + myn];
                by = U[(size_t)(k + 1) * TRAIN + myn];
            }
            bfrag[kb].x = bx;
            bfrag[kb].y = by;
        }

        #pragma unroll 1
        for (int mt = 0; mt < 16; ++mt) {
            v8f acc = {0.f, 0.f, 0.f, 0.f, 0.f, 0.f, 0.f, 0.f};
            #pragma unroll
            for (int kb = 0; kb < 16; ++kb) {
                int row = mt * 16 + nlocal;       // batch row (A matrix M index)
                int kc  = kb * 4 + krow;
                v2f a;
                a.x = su[row * 65 + kc];
                a.y = su[row * 65 + kc + 1];
                acc = __builtin_amdgcn_wmma_f32_16x16x4_f32(
                        false, a, false, bfrag[kb], (short)0, acc, false, false);
            }
            #pragma unroll
            for (int r = 0; r < 8; ++r) {
                int   j  = mt * 16 + r + halfsel * 8;      // C/D layout: M = r (+8 for hi half)
                float ip = 0.5f * acc[r];
                ip = fminf(50.f, fmaxf(-100.f, ip));
                float x = (bl == slab[j]) ? (1.f - ip) : (1.f + ip);   // M = 1
                partial += __logf(1.f + __expf(x));
            }
        }
    }

    red[tid] = partial;
    __syncthreads();
    for (int s = 128; s > 0; s >>= 1) {
        if (tid < s) red[tid] += red[tid + s];
        __syncthreads();
    }
    if (tid == 0) metricP[blockIdx.x] = red[0];
}

__global__ void k_final(const float* __restrict__ metricP, const float* __restrict__ quantP,
                        float* __restrict__ out) {
    if (threadIdx.x == 0 && blockIdx.x == 0) {
        double m = 0.0;
        for (int i = 0; i < NBLK; ++i) m += (double)metricP[i];
        double q = 0.0;
        for (int i = 0; i < NBIT; ++i) q += (double)quantP[i];
        double loss = m / ((double)BATCH * (double)TRAIN)
                    + 0.5 * (q / ((double)NBIT * (double)BATCH));   // ETA = 0.5
        out[0] = (float)loss;
    }
}

// ---------------- launch ----------------

extern "C" void kernel_launch(void* const* d_in, const int* in_sizes, int n_in,
                              void* d_out, int out_size, void* d_ws, size_t ws_size,
                              hipStream_t stream) {
    const float* u   = (const float*)d_in[0];   // (256,64)
    const float* y   = (const float*)d_in[1];   // (256,100)
    const int*   ind = (const int*)  d_in[2];   // (256,)
    const float* U   = (const float*)d_in[3];   // (64,200000)
    const float* Y   = (const float*)d_in[4];   // (100,200000)
    const float* C   = (const float*)d_in[5];   // (64,100)
    float* out = (float*)d_out;

    // workspace layout
    int*   mark    = (int*)d_ws;                       // 200000
    int*   blabel  = mark + TRAIN;                     // 200000
    int*   label   = blabel + TRAIN;                   // 256
    float* b       = (float*)(label + BATCH);          // 64*256
    float* Q       = b + NBIT * BATCH;                 // 100*64
    float* G       = Q + NCLASS * NBIT;                // 100*100
    float* metricP = G + NCLASS * NCLASS;              // NBLK
    float* quantP  = metricP + NBLK;                   // 64

    k_init_mark<<<(TRAIN + 255) / 256, 256, 0, stream>>>(mark);
    k_labels<<<1, 256, 0, stream>>>(y, label);
    k_blabel<<<(TRAIN + 255) / 256, 256, 0, stream>>>(Y, blabel);
    k_scatter<<<1, 256, 0, stream>>>(ind, label, mark, blabel);
    k_bq<<<NBIT, 256, 0, stream>>>(C, u, y, b, quantP);
    k_Q<<<(NCLASS * NBIT + 255) / 256, 256, 0, stream>>>(y, b, Q);
    k_G<<<(NCLASS * NCLASS + 255) / 256, 256, 0, stream>>>(y, G);
    k_dcc<<<1, NBIT, 0, stream>>>(C, Q, G, out);
    k_gemm_loss<<<NBLK, 256, 0, stream>>>(u, U, mark, blabel, label, metricP);
    k_final<<<1, 1, 0, stream>>>(metricP, quantP, out);
}